// HNHNConv_18348100288551
// MI455X (gfx1250) — compile-verified
//
#include <hip/hip_runtime.h>
#include <hip/hip_bf16.h>

// ---------------------------------------------------------------------------
// HNHN hypergraph conv layer for MI455X (gfx1250, wave32, WMMA).
//   Xp = X @ Wv2e + b        (WMMA f32 16x16x4 GEMM)
//   Y  = relu(seg_mean(Xp[v_ids] -> e_ids))
//   Yp = Y @ We2v + b2       (WMMA f32 GEMM)
//   out= relu(seg_mean(Yp[e_ids] -> v_ids))
// Bandwidth-bound (≈3.3 GB scatter/gather); fp32 WMMA keeps precision free.
// ---------------------------------------------------------------------------

typedef __attribute__((ext_vector_type(2))) float v2f;
typedef __attribute__((ext_vector_type(8))) float v8f;

#define C_DIM 128
#define N_EDGES_C 50000

// ---------------------------------------------------------------------------
// Zero-fill (workspace + output accumulators; harness poisons with 0xAA).
// ---------------------------------------------------------------------------
__global__ void hnhn_zero_kernel(float* __restrict__ p, long long n) {
    long long i = (long long)blockIdx.x * blockDim.x + threadIdx.x;
    long long stride = (long long)gridDim.x * blockDim.x;
    for (; i < n; i += stride) p[i] = 0.0f;
}

// ---------------------------------------------------------------------------
// GEMM: Out[M,128] = A[M,128] @ W[128,128] + bias[128]
// One block = 16 output rows; 8 waves, each wave owns a 16x16 N-tile.
// K accumulated with 32 x v_wmma_f32_16x16x4_f32.
// A strip staged in LDS with padded stride (132) -> conflict-free col reads.
// ---------------------------------------------------------------------------
__global__ __launch_bounds__(256) void hnhn_gemm_bias_kernel(
    const float* __restrict__ A, const float* __restrict__ W,
    const float* __restrict__ bias, float* __restrict__ Out, int M)
{
    __shared__ float lds[16 * 132];

    const int tid  = threadIdx.x;
    const int m0   = blockIdx.x * 16;
    if (m0 >= M) return;

    // Stage 16x128 A-strip: thread t -> row t>>4, 8 cols at (t&15)*8.
    {
        const int r = tid >> 4;
        const int c = (tid & 15) * 8;
        const float*  srow = A + (size_t)(m0 + r) * C_DIM + c;
        float*        drow = lds + r * 132 + c;
        *(float4*)(drow + 0) = *(const float4*)(srow + 0);
        *(float4*)(drow + 4) = *(const float4*)(srow + 4);
    }
    __syncthreads();

    const int wave = tid >> 5;        // 8 waves -> 8 N-tiles of 16
    const int lane = tid & 31;
    const int n0   = wave * 16;
    const int half = lane >> 4;       // 0: K=k..k+1, 1: K=k+2..k+3
    const int sub  = lane & 15;       // M index (A) / N index (B)

    v8f acc = {};
#pragma unroll 4
    for (int k = 0; k < C_DIM; k += 4) {
        const int ka = k + half * 2;
        // A fragment: 16x4, lane-half selects K pair (ISA 32-bit A layout)
        v2f a;
        a.x = lds[sub * 132 + ka];
        a.y = lds[sub * 132 + ka + 1];
        // B fragment: 4x16, rows striped across lanes within a VGPR
        v2f b;
        b.x = W[(size_t)ka * C_DIM + n0 + sub];
        b.y = W[(size_t)(ka + 1) * C_DIM + n0 + sub];
        acc = __builtin_amdgcn_wmma_f32_16x16x4_f32(
            /*neg_a=*/false, a, /*neg_b=*/false, b,
            /*c_mod=*/(short)0, acc, /*reuse_a=*/false, /*reuse_b=*/false);
    }

    // C/D layout: VGPR i -> M = i + half*8, N = sub (within tile)
    const float bv = bias[n0 + sub];
#pragma unroll
    for (int i = 0; i < 8; ++i) {
        const int row = m0 + i + half * 8;
        Out[(size_t)row * C_DIM + n0 + sub] = acc[i] + bv;
    }
}

// ---------------------------------------------------------------------------
// Scatter-add one incidence list: dsum[dst] += feats[src], dcnt[dst] += 1.
// Two incidence rows per 256-thread block; thread = one channel (coalesced).
// ---------------------------------------------------------------------------
__global__ __launch_bounds__(256) void hnhn_scatter_kernel(
    const float* __restrict__ feats, const int* __restrict__ src_ids,
    const int* __restrict__ dst_ids, float* __restrict__ dsum,
    float* __restrict__ dcnt, int nnz)
{
    const int i  = blockIdx.x * 2 + (threadIdx.x >> 7);
    const int ch = threadIdx.x & (C_DIM - 1);
    if (i >= nnz) return;
    const int s = src_ids[i];
    const int d = dst_ids[i];
    const float v = feats[(size_t)s * C_DIM + ch];
    atomicAdd(&dsum[(size_t)d * C_DIM + ch], v);
    if (ch == 0) atomicAdd(&dcnt[d], 1.0f);
}

// ---------------------------------------------------------------------------
// In-place mean + ReLU: data[r][c] = relu(data[r][c] / max(cnt[r], 1))
// ---------------------------------------------------------------------------
__global__ __launch_bounds__(256) void hnhn_mean_relu_kernel(
    float* __restrict__ data, const float* __restrict__ cnt, int rows)
{
    const int idx = blockIdx.x * blockDim.x + threadIdx.x;
    if (idx >= rows * C_DIM) return;
    const int r = idx >> 7;
    const float c = fmaxf(cnt[r], 1.0f);
    data[idx] = fmaxf(data[idx] / c, 0.0f);
}

// ---------------------------------------------------------------------------
// Launch
//   d_in: X[V*128], W_v2e[128*128], b_v2e[128], W_e2v[128*128], b_e2v[128],
//         v_ids[NNZ] (int32), e_ids[NNZ] (int32)
//   d_out: relu(Xo) [V*128] fp32
// ---------------------------------------------------------------------------
extern "C" void kernel_launch(void* const* d_in, const int* in_sizes, int n_in,
                              void* d_out, int out_size, void* d_ws, size_t ws_size,
                              hipStream_t stream) {
    const float* X     = (const float*)d_in[0];
    const float* W_v2e = (const float*)d_in[1];
    const float* b_v2e = (const float*)d_in[2];
    const float* W_e2v = (const float*)d_in[3];
    const float* b_e2v = (const float*)d_in[4];
    const int*   v_ids = (const int*)d_in[5];
    const int*   e_ids = (const int*)d_in[6];

    const int V   = in_sizes[0] / C_DIM;   // 100000
    const int NNZ = in_sizes[5];           // 1600000
    const int E   = N_EDGES_C;             // 50000

    // Workspace layout (floats): Xp[V*128] | esum/Y[E*128] | Yp[E*128] | ecnt[E] | vcnt[V]
    float* Xp   = (float*)d_ws;
    float* esum = Xp   + (size_t)V * C_DIM;
    float* Yp   = esum + (size_t)E * C_DIM;
    float* ecnt = Yp   + (size_t)E * C_DIM;
    float* vcnt = ecnt + (size_t)E;
    float* out  = (float*)d_out;           // doubles as vertex-sum accumulator

    // 1. Zero accumulators (esum; ecnt+vcnt are contiguous; out).
    hnhn_zero_kernel<<<2048, 256, 0, stream>>>(esum, (long long)E * C_DIM);
    hnhn_zero_kernel<<<2048, 256, 0, stream>>>(ecnt, (long long)(E + V));
    hnhn_zero_kernel<<<2048, 256, 0, stream>>>(out,  (long long)V * C_DIM);

    // 2. Xp = X @ W_v2e + b_v2e
    hnhn_gemm_bias_kernel<<<(V + 15) / 16, 256, 0, stream>>>(X, W_v2e, b_v2e, Xp, V);

    // 3. v2e scatter: esum[e] += Xp[v], ecnt[e] += 1
    hnhn_scatter_kernel<<<(NNZ + 1) / 2, 256, 0, stream>>>(Xp, v_ids, e_ids, esum, ecnt, NNZ);

    // 4. Y = relu(esum / max(ecnt,1))   (in place -> esum is now Y)
    hnhn_mean_relu_kernel<<<(E * C_DIM + 255) / 256, 256, 0, stream>>>(esum, ecnt, E);

    // 5. Yp = Y @ W_e2v + b_e2v
    hnhn_gemm_bias_kernel<<<(E + 15) / 16, 256, 0, stream>>>(esum, W_e2v, b_e2v, Yp, E);

    // 6. e2v scatter: out[v] += Yp[e], vcnt[v] += 1
    hnhn_scatter_kernel<<<(NNZ + 1) / 2, 256, 0, stream>>>(Yp, e_ids, v_ids, out, vcnt, NNZ);

    // 7. out = relu(out / max(vcnt,1))
    hnhn_mean_relu_kernel<<<(V * C_DIM + 255) / 256, 256, 0, stream>>>(out, vcnt, V);
}